// FixedPointerModel_64201171141027
// MI455X (gfx1250) — compile-verified
//
#include <hip/hip_runtime.h>
#include <hip/hip_bf16.h>
#include <math.h>

typedef __attribute__((ext_vector_type(2))) float v2f;
typedef __attribute__((ext_vector_type(8))) float v8f;

#define B_ 256
#define T_ 512
#define I_ 64
#define D_ 512
#define O_ 128
#define LN_EPS 1e-5f

// Branch-free tanh: hardware v_tanh_f32 if the builtin exists, else
// exp2-based identity tanh(x) = (e^2x - 1)/(e^2x + 1), clamped so the
// exponential never overflows (no inf/inf NaN), using v_exp_f32 + v_rcp_f32.
__device__ __forceinline__ float fast_tanh(float x) {
#if __has_builtin(__builtin_amdgcn_tanhf)
  return __builtin_amdgcn_tanhf(x);
#else
  float xc = fminf(fmaxf(x, -15.0f), 15.0f);
  float t  = __builtin_amdgcn_exp2f(xc * 2.88539008177793f); // 2*log2(e)
  return (t - 1.0f) * __builtin_amdgcn_rcpf(t + 1.0f);
#endif
}

// ---------------------------------------------------------------------------
// Kernel 1: e[bt, d] = tanh( sum_i x[bt,i] * W_in[d,i] + b_in[d] )
// One wave per 16x16 tile, V_WMMA_F32_16X16X4_F32, K-loop over I=64.
// e stores are non-temporal: 256 MB stream, don't thrash the 192 MB L2.
// ---------------------------------------------------------------------------
__global__ __launch_bounds__(256) void embed_gemm_kernel(
    const float* __restrict__ x,     // [B*T, I]
    const float* __restrict__ W_in,  // [D, I]
    const float* __restrict__ b_in,  // [D]
    float* __restrict__ e)           // [B*T, D]
{
  const int lane = threadIdx.x & 31;
  const int wave = threadIdx.x >> 5;
  const int gw   = blockIdx.x * 8 + wave;
  const int m0   = (gw >> 5) * 16;   // BT tile: 0..8191
  const int n0   = (gw & 31) * 16;   // D tile : 0..31

  const int row  = lane & 15;
  const int ksel = (lane >> 4) << 1; // 0 or 2

  const float* aBase = x    + (size_t)(m0 + row) * I_ + ksel;
  const float* bBase = W_in + (size_t)(n0 + row) * I_ + ksel;

  v8f c = {};
#pragma unroll
  for (int k0 = 0; k0 < I_; k0 += 4) {
    v2f a = *(const v2f*)(aBase + k0);
    v2f b = *(const v2f*)(bBase + k0);
    c = __builtin_amdgcn_wmma_f32_16x16x4_f32(
        /*neg_a=*/false, a, /*neg_b=*/false, b,
        /*c_mod=*/(short)0, c, /*reuse_a=*/false, /*reuse_b=*/false);
  }

  const int col   = lane & 15;
  const int rbase = (lane >> 4) << 3; // 0 or 8
  const float bias = b_in[n0 + col];
#pragma unroll
  for (int r = 0; r < 8; ++r) {
    float v = fast_tanh(c[r] + bias);
    __builtin_nontemporal_store(v, &e[(size_t)(m0 + rbase + r) * D_ + (n0 + col)]);
  }
}

// ---------------------------------------------------------------------------
// Kernel 2: sequential scan over T with fused LayerNorm.
// One block (512 threads = 16 wave32) per batch row; recurrent state in
// registers; one barrier per step via double-buffered LDS partials.
// e loads are non-temporal (one-shot stream).
// ---------------------------------------------------------------------------
__global__ __launch_bounds__(512) void scan_kernel(
    const float* __restrict__ e,      // [B, T, D]
    const float* __restrict__ gamma,  // [D]
    const float* __restrict__ beta,   // [D]
    const float* __restrict__ ctx,    // scalar context_strength
    float* __restrict__ hfinal)       // [B, D]
{
  const int b    = blockIdx.x;
  const int d    = threadIdx.x;
  const int wave = d >> 5;
  const int lane = d & 31;

  __shared__ __align__(16) float2 partial[2][16];

  const float scale = 1.0f / (1.0f + __expf(-ctx[0])); // sigmoid
  const float g   = gamma[d];
  const float be  = beta[d];
  const float inv = 1.0f / (float)D_;

  const float* eb = e + (size_t)b * T_ * D_ + d;

  float s_prev = 0.0f, h_prev = 0.0f;
  float e_cur = __builtin_nontemporal_load(eb);

  for (int t = 0; t < T_; ++t) {
    // prefetch next step's element before the reduction/barrier
    float e_next = (t + 1 < T_)
        ? __builtin_nontemporal_load(eb + (size_t)(t + 1) * D_)
        : 0.0f;

    float s = fast_tanh(e_cur + scale * s_prev + h_prev);

    // wave32 butterfly reduction of (sum, sumsq)
    float sum = s, sq = s * s;
#pragma unroll
    for (int off = 16; off >= 1; off >>= 1) {
      sum += __shfl_xor(sum, off, 32);
      sq  += __shfl_xor(sq,  off, 32);
    }
    if (lane == 0) partial[t & 1][wave] = make_float2(sum, sq);
    __syncthreads();

    // cross-wave reduce: 8 x b128 broadcast loads (2 partials each)
    float tsum = 0.0f, tsq = 0.0f;
    const float4* p4 = (const float4*)&partial[t & 1][0];
#pragma unroll
    for (int w = 0; w < 8; ++w) {
      float4 p = p4[w];
      tsum += p.x + p.z;
      tsq  += p.y + p.w;
    }
    float mu  = tsum * inv;
    float var = tsq * inv - mu * mu;
    float h   = (s - mu) * rsqrtf(var + LN_EPS) * g + be;

    s_prev = s;
    h_prev = h;
    e_cur  = e_next;
  }

  hfinal[(size_t)b * D_ + d] = h_prev;
}

// ---------------------------------------------------------------------------
// Kernel 3: logits = hfinal @ W_out^T + b_out   ([256x512]x[512x128])
// ---------------------------------------------------------------------------
__global__ __launch_bounds__(256) void out_gemm_kernel(
    const float* __restrict__ hfinal, // [B, D]
    const float* __restrict__ W_out,  // [O, D]
    const float* __restrict__ b_out,  // [O]
    float* __restrict__ out)          // [B, O]
{
  const int lane = threadIdx.x & 31;
  const int wave = threadIdx.x >> 5;
  const int gw   = blockIdx.x * 8 + wave; // 0..127
  const int m0   = (gw >> 3) * 16;        // B tile
  const int n0   = (gw & 7) * 16;         // O tile

  const int row  = lane & 15;
  const int ksel = (lane >> 4) << 1;

  const float* aBase = hfinal + (size_t)(m0 + row) * D_ + ksel;
  const float* bBase = W_out  + (size_t)(n0 + row) * D_ + ksel;

  v8f c = {};
#pragma unroll 8
  for (int k0 = 0; k0 < D_; k0 += 4) {
    v2f a = *(const v2f*)(aBase + k0);
    v2f b = *(const v2f*)(bBase + k0);
    c = __builtin_amdgcn_wmma_f32_16x16x4_f32(
        false, a, false, b, (short)0, c, false, false);
  }

  const int col   = lane & 15;
  const int rbase = (lane >> 4) << 3;
  const float bias = b_out[n0 + col];
#pragma unroll
  for (int r = 0; r < 8; ++r)
    out[(size_t)(m0 + rbase + r) * O_ + (n0 + col)] = c[r] + bias;
}

// ---------------------------------------------------------------------------
extern "C" void kernel_launch(void* const* d_in, const int* in_sizes, int n_in,
                              void* d_out, int out_size, void* d_ws, size_t ws_size,
                              hipStream_t stream) {
  const float* x     = (const float*)d_in[0];
  const float* W_in  = (const float*)d_in[1];
  const float* b_in  = (const float*)d_in[2];
  const float* gamma = (const float*)d_in[3];
  const float* beta  = (const float*)d_in[4];
  const float* W_out = (const float*)d_in[5];
  const float* b_out = (const float*)d_in[6];
  const float* ctx   = (const float*)d_in[7];
  float* out = (float*)d_out;

  float* e      = (float*)d_ws;                    // B*T*D fp32 = 256 MB
  float* hfinal = e + (size_t)B_ * T_ * D_;        // B*D fp32

  // 1) e = tanh(x @ W_in^T + b_in): (BT/16)*(D/16) tiles, 8 waves/block
  const int tiles1 = (B_ * T_ / 16) * (D_ / 16);   // 262144
  embed_gemm_kernel<<<tiles1 / 8, 256, 0, stream>>>(x, W_in, b_in, e);

  // 2) recurrent scan + layernorm, one block per batch row
  scan_kernel<<<B_, 512, 0, stream>>>(e, gamma, beta, ctx, hfinal);

  // 3) logits = hfinal @ W_out^T + b_out
  const int tiles3 = (B_ / 16) * (O_ / 16);        // 128
  out_gemm_kernel<<<tiles3 / 8, 256, 0, stream>>>(hfinal, W_out, b_out, out);
}